// DAGERC_new_1_81990925681279
// MI455X (gfx1250) — compile-verified
//
#include <hip/hip_runtime.h>
#include <math.h>

// ---------------------------------------------------------------------------
// CDNA5 / gfx1250 implementation of DAGERC forward.
// All heavy math on V_WMMA_F32_16X16X4_F32 (fp32 matrix path = reference dtype).
// ---------------------------------------------------------------------------

typedef __attribute__((ext_vector_type(2))) float v2f;
typedef __attribute__((ext_vector_type(8))) float v8f;

#define DEVI __device__ __forceinline__

static constexpr int Bb  = 8;
static constexpr int Nn  = 256;
static constexpr int Dd  = 256;
static constexpr int Ee  = 1024;
static constexpr int NC  = 7;
static constexpr int IND = 1536;   // 2*D + E

DEVI float waveReduceMax(float v) {
#pragma unroll
  for (int m = 16; m >= 1; m >>= 1) v = fmaxf(v, __shfl_xor(v, m, 32));
  return v;
}
DEVI float waveReduceSum(float v) {
#pragma unroll
  for (int m = 16; m >= 1; m >>= 1) v += __shfl_xor(v, m, 32);
  return v;
}

DEVI v8f wmma_f32(v2f a, v2f b, v8f c) {
  // D = A(16x4,f32) * B(4x16,f32) + C(16x16,f32)
  return __builtin_amdgcn_wmma_f32_16x16x4_f32(false, a, false, b, (short)0, c,
                                               false, false);
}

// ---------------------------------------------------------------------------
// Generic fp32 WMMA GEMM:  C = act(A * op(B) + bias)
//   A: (M,K) row-major, lda.    BMODE==0: B is (K,N) row-major (ldb).
//   BMODE==1: B given as W (N,K) row-major (ldb=K stride) -> op(B)=W^T.
//   One 16x16 tile per wave.  Batched via blockIdx.z with strides.
// ---------------------------------------------------------------------------
template <int BMODE, int ACT>
__global__ void gemm_wmma(const float* __restrict__ A, long lda, long sA,
                          const float* __restrict__ B, long ldb, long sB,
                          float* __restrict__ C, long ldc, long sC,
                          const float* __restrict__ bias,
                          int M, int N, int K) {
  const int wave = threadIdx.x >> 5;
  const int lane = threadIdx.x & 31;
  const int tilesN = (N + 15) >> 4;
  const int tilesM = (M + 15) >> 4;
  const long tile = (long)blockIdx.x * (blockDim.x >> 5) + wave;
  if (tile >= (long)tilesM * tilesN) return;
  const int tm = (int)(tile / tilesN);
  const int tn = (int)(tile % tilesN);
  const float* Ab = A + (long)blockIdx.z * sA;
  const float* Bp = B + (long)blockIdx.z * sB;
  float* Cb = C + (long)blockIdx.z * sC;

  const int m0 = tm * 16, n0 = tn * 16;
  const int mrow = m0 + (lane & 15);
  const int ncol = n0 + (lane & 15);
  const bool mok = (mrow < M);
  const bool nok = (ncol < N);
  const int koff = (lane >> 4) * 2;

  const float* Arow = Ab + (long)(mok ? mrow : m0) * lda + koff;
  const float* Brow;
  if (BMODE == 1) Brow = Bp + (long)(nok ? ncol : n0) * ldb + koff;
  else            Brow = Bp + (nok ? ncol : n0);

  v8f acc = {0.f, 0.f, 0.f, 0.f, 0.f, 0.f, 0.f, 0.f};
  for (int k0 = 0; k0 < K; k0 += 4) {
    v2f a; a.x = Arow[k0]; a.y = Arow[k0 + 1];
    v2f b;
    if (BMODE == 1) { b.x = Brow[k0]; b.y = Brow[k0 + 1]; }
    else {
      b.x = Brow[(long)(k0 + koff) * ldb];
      b.y = Brow[(long)(k0 + koff + 1) * ldb];
    }
    acc = wmma_f32(a, b, acc);
  }
  if (!nok) return;
  const float bv = bias ? bias[ncol] : 0.0f;
  const int mbase = m0 + (lane >> 4) * 8;
#pragma unroll
  for (int v = 0; v < 8; ++v) {
    int mm = mbase + v;
    if (mm < M) {
      float x = acc[v] + bv;
      if (ACT == 1) x = fmaxf(x, 0.0f);
      Cb[(long)mm * ldc + ncol] = x;
    }
  }
}

// ---------------------------------------------------------------------------
// In-scan gate GEMM:  sG(16x768) rows{0..7}=sA_short@Ws^T, rows{8..15}=..@Wl^T
// 48 N-tiles x 2 branches = 96 wave-tasks over 32 waves.
// ---------------------------------------------------------------------------
DEVI void scan_gate_gemm(const float (*sA)[Dd], const float* __restrict__ Ws,
                         const float* __restrict__ Wl, float (*sG)[3 * Dd],
                         int wave, int lane) {
  const int m = lane & 15;
  const int koff = (lane >> 4) * 2;
#pragma unroll
  for (int t = 0; t < 3; ++t) {
    const int task = wave + 32 * t;
    const int brw  = task / 48;            // 0 = short rows, 1 = long rows
    const int n0   = (task % 48) * 16;
    const int ncol = n0 + (lane & 15);
    const float* Wrow = (brw ? Wl : Ws) + (long)ncol * Dd + koff;
    v8f acc = {0.f, 0.f, 0.f, 0.f, 0.f, 0.f, 0.f, 0.f};
    for (int k0 = 0; k0 < Dd; k0 += 4) {
      v2f a; a.x = sA[m][k0 + koff]; a.y = sA[m][k0 + koff + 1];
      v2f b; b.x = Wrow[k0];         b.y = Wrow[k0 + 1];
      acc = wmma_f32(a, b, acc);
    }
    if ((lane >> 4) == brw) {              // C rows v / v+8 per lane half
      const int rbase = brw * 8;
#pragma unroll
      for (int v = 0; v < 8; ++v) sG[rbase + v][ncol] = acc[v];
    }
  }
}

// ---------------------------------------------------------------------------
// Persistent DAG scan kernel: one workgroup (32 wave32 waves), 256 steps.
// Row index r = br*8 + b  (br: 0=short branch, 1=long branch).
// ---------------------------------------------------------------------------
__global__ __launch_bounds__(1024) void dag_scan(
    const float* __restrict__ Hprev_s, const float* __restrict__ Hprev_l,
    float* __restrict__ Hout_s, float* __restrict__ Hout_l,
    const float* __restrict__ adj1, const float* __restrict__ adj2,
    const int* __restrict__ smask,
    const float* __restrict__ CGI_s, const float* __restrict__ CGI_l,
    const float* __restrict__ PGH_s, const float* __restrict__ PGH_l,
    const float* __restrict__ cwhh_s, const float* __restrict__ cwhh_l,
    const float* __restrict__ cbhh_s, const float* __restrict__ cbhh_l,
    const float* __restrict__ pwih_s, const float* __restrict__ pwih_l,
    const float* __restrict__ pbih_s, const float* __restrict__ pbih_l,
    const float* __restrict__ wr0_s, const float* __restrict__ wr0_l,
    const float* __restrict__ wr1_s, const float* __restrict__ wr1_l,
    const float* __restrict__ linw_s, const float* __restrict__ linw_l) {
  __shared__ float sKH[16][Nn];       // 16 KB  key-side attention logits
  __shared__ float sW [16][Nn];       // 16 KB  signed softmax weights
  __shared__ float sU [16][2 * Dd];   // 32 KB  [u0 | u1]
  __shared__ float sM [16][Dd];       // 16 KB  attention message M
  __shared__ float sG [16][3 * Dd];   // 48 KB  gate pre-activations
  __shared__ float sH [16][Dd];       // 16 KB  Cg then H_i

  const int tid  = threadIdx.x;
  const int wave = tid >> 5;
  const int lane = tid & 31;

  for (int i = 0; i < Nn; ++i) {
    // ---- Phase A: masked softmax stats over prefix j<i (1 wave / row) ----
    if (wave < 16) {
      const int row = wave, br = row >> 3, b = row & 7;
      const float* adj = (br ? adj2 : adj1) + ((long)(b * Nn + i)) * Nn;
      const int*   sm  = smask + ((long)(b * Nn + i)) * Nn;
      float mx = -3.0e38f;
      for (int j = lane; j < i; j += 32)
        if (adj[j] != 0.f) mx = fmaxf(mx, sKH[row][j]);
      mx = waveReduceMax(mx);
      float se = 0.f;
      for (int j = lane; j < i; j += 32) {
        float w = (adj[j] != 0.f) ? __expf(sKH[row][j] - mx) : 0.f;
        sW[row][j] = sm[j] ? w : -w;   // sign encodes s_mask
        se += w;
      }
      se = waveReduceSum(se);
      const float inv = (se > 0.f) ? 1.f / se : 0.f;
      for (int j = lane; j < i; j += 32) sW[row][j] *= inv;
    }
    __syncthreads();

    // ---- Phase A2: u0/u1 weighted prefix sums (2 waves / row) ----
    {
      const int row = wave >> 1;
      const int d0  = (wave & 1) * 128 + lane * 4;
      const float* Hrow =
          (row < 8 ? Hout_s : Hout_l) + ((long)(row & 7)) * Nn * Dd;
      float u0[4] = {0, 0, 0, 0}, u1[4] = {0, 0, 0, 0};
      for (int j = 0; j < i; ++j) {
        const float sw = sW[row][j];
        const float w0 = fmaxf(sw, 0.f), w1 = fmaxf(-sw, 0.f);
        const float4 h = *(const float4*)(Hrow + (long)j * Dd + d0);
        u0[0] += w0 * h.x; u0[1] += w0 * h.y; u0[2] += w0 * h.z; u0[3] += w0 * h.w;
        u1[0] += w1 * h.x; u1[1] += w1 * h.y; u1[2] += w1 * h.z; u1[3] += w1 * h.w;
      }
#pragma unroll
      for (int c = 0; c < 4; ++c) {
        sU[row][d0 + c]      = u0[c];
        sU[row][Dd + d0 + c] = u1[c];
      }
    }
    __syncthreads();

    // ---- GEMM1: M = u0@wr0^T + u1@wr1^T (branch-separate B) ----
    {
      const int brw = wave >> 4;
      const int n0  = (wave & 15) * 16;
      const float* W0 = brw ? wr0_l : wr0_s;
      const float* W1 = brw ? wr1_l : wr1_s;
      const int m = lane & 15;
      const int ncol = n0 + (lane & 15);
      const int koff = (lane >> 4) * 2;
      v8f acc = {0.f, 0.f, 0.f, 0.f, 0.f, 0.f, 0.f, 0.f};
      for (int k0 = 0; k0 < 2 * Dd; k0 += 4) {
        const int kk = k0 + koff;                 // kk<Dd uniform per wave
        v2f a; a.x = sU[m][kk]; a.y = sU[m][kk + 1];
        const float* Wp = (kk < Dd) ? (W0 + (long)ncol * Dd + kk)
                                    : (W1 + (long)ncol * Dd + (kk - Dd));
        v2f b; b.x = Wp[0]; b.y = Wp[1];
        acc = wmma_f32(a, b, acc);
      }
      if ((lane >> 4) == brw) {
        const int rbase = brw * 8;
#pragma unroll
        for (int v = 0; v < 8; ++v) sM[rbase + v][ncol] = acc[v];
      }
    }
    __syncthreads();

    // ---- GEMM2: gh_c = M @ c_whh^T ----
    scan_gate_gemm(sM, cwhh_s, cwhh_l, sG, wave, lane);
    __syncthreads();

    // ---- Phase C1: Cg = GRU(x=x_i (precomp), h=M) ----
    {
      const int row = tid >> 6;
      const int d   = (tid & 63) * 4;
      const int br = row >> 3, b = row & 7;
      const float* gi = (br ? CGI_l : CGI_s) + ((long)(b * Nn + i)) * 768;
      const float* bh = br ? cbhh_l : cbhh_s;
#pragma unroll
      for (int c = 0; c < 4; ++c) {
        const int dd = d + c;
        const float ir = gi[dd], iz = gi[Dd + dd], in_ = gi[2 * Dd + dd];
        const float hr = sG[row][dd] + bh[dd];
        const float hz = sG[row][Dd + dd] + bh[Dd + dd];
        const float hn = sG[row][2 * Dd + dd] + bh[2 * Dd + dd];
        const float r = 1.f / (1.f + __expf(-(ir + hr)));
        const float z = 1.f / (1.f + __expf(-(iz + hz)));
        const float n = tanhf(in_ + r * hn);
        sH[row][dd] = (1.f - z) * n + z * sM[row][dd];
      }
    }
    __syncthreads();

    // ---- GEMM3: gi_p = M @ p_wih^T ----
    scan_gate_gemm(sM, pwih_s, pwih_l, sG, wave, lane);
    __syncthreads();

    // ---- Phase C2: Pg = GRU(x=M, h=x_i);  H_i = Cg + Pg ----
    {
      const int row = tid >> 6;
      const int d   = (tid & 63) * 4;
      const int br = row >> 3, b = row & 7;
      const float* gh = (br ? PGH_l : PGH_s) + ((long)(b * Nn + i)) * 768;
      const float* bi = br ? pbih_l : pbih_s;
      const float* xp = (br ? Hprev_l : Hprev_s) + ((long)(b * Nn + i)) * Dd;
      float* Ho = (br ? Hout_l : Hout_s) + ((long)(b * Nn + i)) * Dd;
#pragma unroll
      for (int c = 0; c < 4; ++c) {
        const int dd = d + c;
        const float ir = sG[row][dd] + bi[dd];
        const float iz = sG[row][Dd + dd] + bi[Dd + dd];
        const float in_ = sG[row][2 * Dd + dd] + bi[2 * Dd + dd];
        const float r = 1.f / (1.f + __expf(-(ir + gh[dd])));
        const float z = 1.f / (1.f + __expf(-(iz + gh[Dd + dd])));
        const float n = tanhf(in_ + r * gh[2 * Dd + dd]);
        const float xi = xp[dd];
        const float Hv = sH[row][dd] + (1.f - z) * n + z * xi;
        sH[row][dd] = Hv;
        Ho[dd] = Hv;
      }
    }
    __syncthreads();

    // ---- KH[i] = H_i . lin_w[D:2D] ----
    if (wave < 16) {
      const int row = wave, br = row >> 3;
      const float* lw = (br ? linw_l : linw_s) + Dd;
      float p = 0.f;
      for (int d = lane; d < Dd; d += 32) p += sH[row][d] * lw[d];
      p = waveReduceSum(p);
      if (lane == 0) sKH[row][i] = p;
    }
    __syncthreads();
  }
}

// ---------------------------------------------------------------------------
// Row softmax (one wave per row).
// ---------------------------------------------------------------------------
__global__ void softmax_rows(float* __restrict__ X, int rows, int cols) {
  const int wave = threadIdx.x >> 5, lane = threadIdx.x & 31;
  const int row = blockIdx.x * (blockDim.x >> 5) + wave;
  if (row >= rows) return;
  float* p = X + (long)row * cols;
  float mx = -3.0e38f;
  for (int c = lane; c < cols; c += 32) mx = fmaxf(mx, p[c]);
  mx = waveReduceMax(mx);
  float s = 0.f;
  for (int c = lane; c < cols; c += 32) { float e = __expf(p[c] - mx); p[c] = e; s += e; }
  s = waveReduceSum(s);
  const float inv = 1.f / s;
  for (int c = lane; c < cols; c += 32) p[c] *= inv;
}

// ---------------------------------------------------------------------------
// Hf = concat(HSn, HLn, features) along feature dim.
// ---------------------------------------------------------------------------
__global__ void concat_hf(const float* __restrict__ HSn,
                          const float* __restrict__ HLn,
                          const float* __restrict__ feat,
                          float* __restrict__ Hf) {
  const long total = (long)Bb * Nn * IND;
  for (long idx = (long)blockIdx.x * blockDim.x + threadIdx.x; idx < total;
       idx += (long)gridDim.x * blockDim.x) {
    const long m = idx / IND;
    const int c = (int)(idx % IND);
    float v;
    if (c < Dd)           v = HSn[m * Dd + c];
    else if (c < 2 * Dd)  v = HLn[m * Dd + (c - Dd)];
    else                  v = feat[m * Ee + (c - 2 * Dd)];
    Hf[idx] = v;
  }
}

// ---------------------------------------------------------------------------
// Diff-loss: per-block partial centered Grams (36 upper pairs of 8x8).
// ---------------------------------------------------------------------------
__global__ void diff_partial(const float* __restrict__ X,
                             float* __restrict__ out /* [gridDim.x][36] */) {
  constexpr long F = (long)Nn * Dd;  // 65536
  const long cpb = F / gridDim.x;
  const long c0 = (long)blockIdx.x * cpb;
  float acc[36];
#pragma unroll
  for (int p = 0; p < 36; ++p) acc[p] = 0.f;
  for (long c = c0 + threadIdx.x; c < c0 + cpb; c += blockDim.x) {
    float v[8]; float mean = 0.f;
#pragma unroll
    for (int b = 0; b < 8; ++b) { v[b] = X[b * F + c]; mean += v[b]; }
    mean *= 0.125f;
#pragma unroll
    for (int b = 0; b < 8; ++b) v[b] -= mean;
    int p = 0;
#pragma unroll
    for (int i = 0; i < 8; ++i)
#pragma unroll
      for (int j = i; j < 8; ++j) { acc[p] += v[i] * v[j]; ++p; }
  }
  __shared__ float red[256];
  for (int p = 0; p < 36; ++p) {
    red[threadIdx.x] = acc[p];
    __syncthreads();
    for (int s = blockDim.x >> 1; s > 0; s >>= 1) {
      if ((int)threadIdx.x < s) red[threadIdx.x] += red[threadIdx.x + s];
      __syncthreads();
    }
    if (threadIdx.x == 0) out[(long)blockIdx.x * 36 + p] = red[0];
    __syncthreads();
  }
}

__global__ void diff_final(const float* __restrict__ Ps0,
                           const float* __restrict__ Pl0,
                           const float* __restrict__ Ps1,
                           const float* __restrict__ Pl1, int nb,
                           float* __restrict__ out) {
  __shared__ float G[4][36];
  for (int q = threadIdx.x; q < 4 * 36; q += blockDim.x) {
    const int w = q / 36, p = q % 36;
    const float* P = (w == 0) ? Ps0 : (w == 1) ? Pl0 : (w == 2) ? Ps1 : Pl1;
    float s = 0.f;
    for (int b = 0; b < nb; ++b) s += P[b * 36 + p];
    G[w][p] = s;
  }
  __syncthreads();
  if (threadIdx.x == 0) {
    const float denom = 65536.0f * 65536.0f;
    float dl = 0.f;
    for (int layer = 0; layer < 2; ++layer) {
      const float* G1 = G[layer * 2 + 0];
      const float* G2 = G[layer * 2 + 1];
      float n1[8], n2[8];
      {
        int p = 0;
        for (int i = 0; i < 8; ++i) {
          n1[i] = sqrtf(fmaxf(G1[p], 0.f)) + 1e-6f;
          n2[i] = sqrtf(fmaxf(G2[p], 0.f)) + 1e-6f;
          p += (8 - i);  // advance past row i's pairs (i,i)..(i,7)
        }
      }
      int p = 0;
      for (int i = 0; i < 8; ++i)
        for (int j = i; j < 8; ++j, ++p) {
          const float g1 = G1[p] / (n1[i] * n1[j]);
          const float g2 = G2[p] / (n2[i] * n2[j]);
          dl += ((i == j) ? 1.f : 2.f) * g1 * g2 / denom;
        }
    }
    out[0] = 0.3f * dl;  // BETA * (dl0 + dl1)
  }
}

// ---------------------------------------------------------------------------
// Host side
// ---------------------------------------------------------------------------
struct BranchP {
  const float *c_bhh, *c_bih, *c_whh, *c_wih, *lin_b, *lin_w;
  const float *p_bhh, *p_bih, *p_whh, *p_wih, *wr0, *wr1;
};
static BranchP getBP(void* const* din, int base) {
  BranchP p;
  p.c_bhh = (const float*)din[base + 0];  p.c_bih = (const float*)din[base + 1];
  p.c_whh = (const float*)din[base + 2];  p.c_wih = (const float*)din[base + 3];
  p.lin_b = (const float*)din[base + 4];  p.lin_w = (const float*)din[base + 5];
  p.p_bhh = (const float*)din[base + 6];  p.p_bih = (const float*)din[base + 7];
  p.p_whh = (const float*)din[base + 8];  p.p_wih = (const float*)din[base + 9];
  p.wr0   = (const float*)din[base + 10]; p.wr1   = (const float*)din[base + 11];
  return p;
}

static void gemmL(hipStream_t st, const float* A, long lda, long sA,
                  const float* B, long ldb, long sB, float* C, long ldc,
                  long sC, const float* bias, int M, int N, int K, int batch,
                  int bmode, int act) {
  const int tiles = ((M + 15) / 16) * ((N + 15) / 16);
  dim3 grid((tiles + 7) / 8, 1, batch), block(256);
  if (bmode == 1 && act == 1)
    gemm_wmma<1, 1><<<grid, block, 0, st>>>(A, lda, sA, B, ldb, sB, C, ldc, sC, bias, M, N, K);
  else if (bmode == 1)
    gemm_wmma<1, 0><<<grid, block, 0, st>>>(A, lda, sA, B, ldb, sB, C, ldc, sC, bias, M, N, K);
  else if (act == 1)
    gemm_wmma<0, 1><<<grid, block, 0, st>>>(A, lda, sA, B, ldb, sB, C, ldc, sC, bias, M, N, K);
  else
    gemm_wmma<0, 0><<<grid, block, 0, st>>>(A, lda, sA, B, ldb, sB, C, ldc, sC, bias, M, N, K);
}

extern "C" void kernel_launch(void* const* d_in, const int* in_sizes, int n_in,
                              void* d_out, int out_size, void* d_ws,
                              size_t ws_size, hipStream_t stream) {
  (void)in_sizes; (void)n_in; (void)out_size; (void)ws_size;
  // Flattening order: dict insertion at top level, params pytree sorted-by-key.
  const float* features = (const float*)d_in[0];
  const float* adj1     = (const float*)d_in[1];
  const float* adj2     = (const float*)d_in[2];
  const int*   smask    = (const int*)d_in[3];
  const float* affine1  = (const float*)d_in[6];
  const float* affine2  = (const float*)d_in[7];
  const float* fc1_b    = (const float*)d_in[8];
  const float* fc1_w    = (const float*)d_in[9];
  BranchP L0 = getBP(d_in, 10), L1 = getBP(d_in, 22);
  const float* mlp_b0 = (const float*)d_in[34];
  const float* mlp_b1 = (const float*)d_in[35];
  const float* mlp_w0 = (const float*)d_in[36];
  const float* mlp_w1 = (const float*)d_in[37];
  const float* out_b  = (const float*)d_in[38];
  const float* out_w  = (const float*)d_in[39];
  BranchP S0 = getBP(d_in, 40), S1 = getBP(d_in, 52);

  float* ws = (float*)d_ws;
  size_t off = 0;
  auto take = [&](size_t n) { float* p = ws + off; off += n; return p; };
  const size_t MN = (size_t)Bb * Nn;            // 2048
  float* H0   = take(MN * Dd);
  float* Hs0  = take(MN * Dd);
  float* Hl0  = take(MN * Dd);
  float* Hs1  = take(MN * Dd);
  float* Hl1  = take(MN * Dd);
  float* CGIs = take(MN * 768);
  float* CGIl = take(MN * 768);
  float* PGHs = take(MN * 768);
  float* PGHl = take(MN * 768);
  float* Tbuf = take(MN * Dd);
  float* Abuf = take(MN * Dd);
  float* HSn  = take(MN * Dd);
  float* HLn  = take(MN * Dd);
  float* Hf   = take(MN * IND);
  float* Xbuf = take(MN * Dd);
  float* P0s = take(64 * 36);
  float* P0l = take(64 * 36);
  float* P1s = take(64 * 36);
  float* P1l = take(64 * 36);

  // H0 = relu(features @ fc1_w^T + fc1_b)
  gemmL(stream, features, Ee, 0, fc1_w, Ee, 0, H0, Dd, 0, fc1_b,
        (int)MN, Dd, Ee, 1, /*bmode*/1, /*relu*/1);

  // Layer loop: precompute GRU x-side GEMMs, then persistent scan.
  for (int l = 0; l < 2; ++l) {
    const float* Hps = l ? Hs0 : H0;
    const float* Hpl = l ? Hl0 : H0;
    float* Hos = l ? Hs1 : Hs0;
    float* Hol = l ? Hl1 : Hl0;
    const BranchP& S = l ? S1 : S0;
    const BranchP& Lg = l ? L1 : L0;

    gemmL(stream, Hps, Dd, 0, S.c_wih, Dd, 0, CGIs, 768, 0, S.c_bih,
          (int)MN, 768, Dd, 1, 1, 0);
    gemmL(stream, Hpl, Dd, 0, Lg.c_wih, Dd, 0, CGIl, 768, 0, Lg.c_bih,
          (int)MN, 768, Dd, 1, 1, 0);
    gemmL(stream, Hps, Dd, 0, S.p_whh, Dd, 0, PGHs, 768, 0, S.p_bhh,
          (int)MN, 768, Dd, 1, 1, 0);
    gemmL(stream, Hpl, Dd, 0, Lg.p_whh, Dd, 0, PGHl, 768, 0, Lg.p_bhh,
          (int)MN, 768, Dd, 1, 1, 0);

    dag_scan<<<1, 1024, 0, stream>>>(
        Hps, Hpl, Hos, Hol, adj1, adj2, smask, CGIs, CGIl, PGHs, PGHl,
        S.c_whh, Lg.c_whh, S.c_bhh, Lg.c_bhh, S.p_wih, Lg.p_wih, S.p_bih,
        Lg.p_bih, S.wr0, Lg.wr0, S.wr1, Lg.wr1, S.lin_w, Lg.lin_w);
  }

  // Diff-loss partial Grams (deterministic two-stage reduction).
  diff_partial<<<64, 256, 0, stream>>>(Hs0, P0s);
  diff_partial<<<64, 256, 0, stream>>>(Hl0, P0l);
  diff_partial<<<64, 256, 0, stream>>>(Hs1, P1s);
  diff_partial<<<64, 256, 0, stream>>>(Hl1, P1l);

  // Cross-channel fusion (HS = Hs1, HL = Hl1).
  const long bstr = (long)Nn * Dd;
  // T = HS @ affine1 ; A1 = softmax(T @ HL^T) ; HSn = A1 @ HL
  gemmL(stream, Hs1, Dd, 0, affine1, Dd, 0, Tbuf, Dd, 0, nullptr,
        (int)MN, Dd, Dd, 1, 0, 0);
  gemmL(stream, Tbuf, Dd, bstr, Hl1, Dd, bstr, Abuf, Nn, (long)Nn * Nn,
        nullptr, Nn, Nn, Dd, Bb, 1, 0);
  softmax_rows<<<(int)MN / 8, 256, 0, stream>>>(Abuf, (int)MN, Nn);
  gemmL(stream, Abuf, Nn, (long)Nn * Nn, Hl1, Dd, bstr, HSn, Dd, bstr,
        nullptr, Nn, Dd, Nn, Bb, 0, 0);
  // T = HL @ affine2 ; A2 = softmax(T @ HS^T) ; HLn = A2 @ HS
  gemmL(stream, Hl1, Dd, 0, affine2, Dd, 0, Tbuf, Dd, 0, nullptr,
        (int)MN, Dd, Dd, 1, 0, 0);
  gemmL(stream, Tbuf, Dd, bstr, Hs1, Dd, bstr, Abuf, Nn, (long)Nn * Nn,
        nullptr, Nn, Nn, Dd, Bb, 1, 0);
  softmax_rows<<<(int)MN / 8, 256, 0, stream>>>(Abuf, (int)MN, Nn);
  gemmL(stream, Abuf, Nn, (long)Nn * Nn, Hs1, Dd, bstr, HLn, Dd, bstr,
        nullptr, Nn, Dd, Nn, Bb, 0, 0);

  // Head.
  concat_hf<<<2048, 256, 0, stream>>>(HSn, HLn, features, Hf);
  gemmL(stream, Hf, IND, 0, mlp_w0, IND, 0, Xbuf, Dd, 0, mlp_b0,
        (int)MN, Dd, IND, 1, 1, 1);
  gemmL(stream, Xbuf, Dd, 0, mlp_w1, Dd, 0, Tbuf, Dd, 0, mlp_b1,
        (int)MN, Dd, Dd, 1, 1, 1);
  gemmL(stream, Tbuf, Dd, 0, out_w, Dd, 0, (float*)d_out, NC, 0, out_b,
        (int)MN, NC, Dd, 1, 1, 0);

  // Scalar output: BETA * (dl0 + dl1) at d_out[B*N*NC].
  diff_final<<<1, 64, 0, stream>>>(P0s, P0l, P1s, P1l, 64,
                                   (float*)d_out + (size_t)Bb * Nn * NC);
}